// ClassCapsule_13116830122362
// MI455X (gfx1250) — compile-verified
//
#include <hip/hip_runtime.h>

typedef __attribute__((ext_vector_type(2))) float v2f;
typedef __attribute__((ext_vector_type(8))) float v8f;

#define CI 32
#define CO 32
#define CH 18
#define NPOS 1024   // B*H*W = 16*8*8
#define XSTRIDE 544 // CI*17

// One block per spatial position n. 256 threads = 8 waves.
// Phase 1: votes[i][c][18] into LDS via v_wmma_f32_16x16x4_f32 (transposed mapping:
//          rows = (c_local,d), K = b, cols = a; only cols 0..3 valid).
// Phase 2: 3 EM-routing iterations entirely in LDS.
// Phase 3: write per-position miu (32*18) and act_out (32) to workspace.
__global__ __launch_bounds__(256) void caps_em_kernel(
    const float* __restrict__ x, const float* __restrict__ w,
    const float* __restrict__ beta_v, const float* __restrict__ beta_a,
    const float* __restrict__ coord,
    float* __restrict__ miu_ws, float* __restrict__ act_ws)
{
    __shared__ float votes_s[CI][CO][CH];   // 73728 B
    __shared__ float r_s[CI][CO];           // 4096 B
    __shared__ float tA[CI][CO];            // 4096 B
    __shared__ float tB[CI][CO];            // 4096 B
    __shared__ float miu_s[CO][CH];         // 2304 B
    __shared__ float sig_s[CO][CH];         // 2304 B
    __shared__ float act_in[CI];
    __shared__ float act_o[CO];
    __shared__ float rsum_s[CO];
    __shared__ float rowsum[CI];
    __shared__ float imax_s[CI];
    __shared__ float asum_s[CI];

    const int tid = threadIdx.x;
    const int n   = blockIdx.x;
    const float* xn = x + (size_t)n * XSTRIDE;

    // ---------------- Phase 1: votes via WMMA f32 16x16x4 ----------------
    {
        const int lane = tid & 31;
        const int wv   = tid >> 5;       // wave id -> c-block (4 output capsules)
        const int lrow = lane & 15;      // row index within 16
        const int lhi  = lane >> 4;      // lane half selects K pair
        const int cA   = wv * 4 + (lrow >> 2);  // output capsule for A row
        const int dA   = lrow & 3;              // d for A row

        for (int i = 0; i < CI; ++i) {
            // A[m][k] = w[i, cA, k, dA], k = 2*lhi + j  (ISA 32-bit A 16x4 layout)
            const float* wi = w + (size_t)i * 512 + (size_t)cA * 16 + dA;
            v2f Av;
            Av.x = wi[(2 * lhi + 0) * 4];
            Av.y = wi[(2 * lhi + 1) * 4];
            // B[k][col] = pose[a=col][b=k] = x[n, i*17 + col*4 + k], valid for col<4
            float b0 = 0.0f, b1 = 0.0f;
            if (lrow < 4) {
                b0 = xn[i * 17 + lrow * 4 + 2 * lhi + 0];
                b1 = xn[i * 17 + lrow * 4 + 2 * lhi + 1];
            }
            v2f Bv; Bv.x = b0; Bv.y = b1;
            v8f acc = {};
            v8f d8 = __builtin_amdgcn_wmma_f32_16x16x4_f32(
                false, Av, false, Bv, (short)0, acc, false, false);
            // D row M = v + 8*lhi -> (c_local = M>>2, d = M&3); col = lrow = a
            if (lrow < 4) {
                const int a = lrow;
                #pragma unroll
                for (int v = 0; v < 8; ++v) {
                    const int M = v + 8 * lhi;
                    votes_s[i][wv * 4 + (M >> 2)][2 + a * 4 + (M & 3)] = d8[v];
                }
            }
        }
    }
    // coord channels, input activations, r init
    {
        const float c0 = coord[(n & 63) * 2 + 0];
        const float c1 = coord[(n & 63) * 2 + 1];
        for (int p = tid; p < CI * CO; p += 256) {
            const int i = p >> 5, c = p & 31;
            votes_s[i][c][0] = c0;
            votes_s[i][c][1] = c1;
            r_s[i][c] = 1.0f / (float)CO;
        }
        if (tid < CI) act_in[tid] = xn[tid * 17 + 16];
    }
    __syncthreads();

    const float EPSf = 1e-9f;
    const float LN10 = 2.302585092994046f;

    // ---------------- Phase 2: EM routing (3 iterations) ----------------
    for (int it = 0; it < 3; ++it) {
        if (it > 0) {
            // E-step: per-(i,c) log-likelihood sum + per-(i,c) max over ch
            for (int p = tid; p < CI * CO; p += 256) {
                const int i = p >> 5, c = p & 31;
                float s = 0.0f, m = -3.0e38f;
                for (int ch = 0; ch < CH; ++ch) {
                    const float sg = sig_s[c][ch];
                    const float dv = votes_s[i][c][ch] - miu_s[c][ch];
                    const float lp = -0.5f * __logf(sg) - dv * dv / (2.0f * sg);
                    s += lp;
                    m = fmaxf(m, lp);
                }
                tA[i][c] = s;
                tB[i][c] = m;
            }
            __syncthreads();
            if (tid < CI) {  // max over (c,ch) per i
                float m = -3.0e38f;
                for (int c = 0; c < CO; ++c) m = fmaxf(m, tB[tid][c]);
                imax_s[tid] = m;
            }
            __syncthreads();
            for (int p = tid; p < CI * CO; p += 256) {
                const int i = p >> 5, c = p & 31;
                // exp(sum_ch(log_p - (max - ln10))) * act_out[c]
                tA[i][c] = __expf(tA[i][c] - 18.0f * (imax_s[i] - LN10)) * act_o[c];
            }
            __syncthreads();
            if (tid < CI) {
                float s = 0.0f;
                for (int c = 0; c < CO; ++c) s += tA[tid][c];
                asum_s[tid] = s;
            }
            __syncthreads();
            for (int p = tid; p < CI * CO; p += 256) {
                const int i = p >> 5, c = p & 31;
                r_s[i][c] = tA[i][c] / (asum_s[i] + EPSf);
            }
            __syncthreads();
        }
        // ---- M-step ----
        for (int p = tid; p < CI * CO; p += 256) {
            const int i = p >> 5, c = p & 31;
            r_s[i][c] *= act_in[i];
        }
        __syncthreads();
        if (tid < CI) {
            float s = 0.0f;
            for (int c = 0; c < CO; ++c) s += r_s[tid][c];
            rowsum[tid] = s;
        }
        __syncthreads();
        for (int p = tid; p < CI * CO; p += 256) {
            const int i = p >> 5, c = p & 31;
            r_s[i][c] /= (rowsum[i] + EPSf);
        }
        __syncthreads();
        if (tid < CO) {
            float s = 0.0f;
            for (int i = 0; i < CI; ++i) s += r_s[i][tid];
            rsum_s[tid] = s;
        }
        __syncthreads();
        for (int q = tid; q < CO * CH; q += 256) {
            const int c = q / CH, ch = q % CH;
            float s = 0.0f;
            for (int i = 0; i < CI; ++i) s += votes_s[i][c][ch] * r_s[i][c];
            miu_s[c][ch] = s / (rsum_s[c] + EPSf);
        }
        __syncthreads();
        for (int q = tid; q < CO * CH; q += 256) {
            const int c = q / CH, ch = q % CH;
            const float mu  = miu_s[c][ch];
            const float inv = 1.0f / (rsum_s[c] + EPSf);
            float s = 0.0f;
            for (int i = 0; i < CI; ++i) {
                const float d = votes_s[i][c][ch] - mu;
                s += d * d * r_s[i][c];
            }
            sig_s[c][ch] = s * inv + EPSf;
        }
        __syncthreads();
        if (it == 2) {
            if (tid < CO) {
                float cost = 0.0f;
                for (int ch = 0; ch < CH; ++ch)
                    cost += beta_v[tid * CH + ch] + 0.5f * __logf(sig_s[tid][ch]);
                cost *= rsum_s[tid];
                rowsum[tid] = 0.03f * (beta_a[tid] - cost);  // logits (reuse rowsum)
            }
            __syncthreads();
            if (tid == 0) {  // softmax over co
                float m = -3.0e38f;
                for (int c = 0; c < CO; ++c) m = fmaxf(m, rowsum[c]);
                float s = 0.0f;
                for (int c = 0; c < CO; ++c) { const float e = __expf(rowsum[c] - m); act_o[c] = e; s += e; }
                const float inv = 1.0f / s;
                for (int c = 0; c < CO; ++c) act_o[c] *= inv;
            }
        } else {
            if (tid == 0) {  // softmax(r_sum) over co
                float m = -3.0e38f;
                for (int c = 0; c < CO; ++c) m = fmaxf(m, rsum_s[c]);
                float s = 0.0f;
                for (int c = 0; c < CO; ++c) { const float e = __expf(rsum_s[c] - m); act_o[c] = e; s += e; }
                const float inv = 1.0f / s;
                for (int c = 0; c < CO; ++c) act_o[c] *= inv;
            }
        }
        __syncthreads();
    }

    // ---------------- Phase 3: write per-position results ----------------
    for (int q = tid; q < CO * CH; q += 256)
        miu_ws[(size_t)n * (CO * CH) + q] = miu_s[q / CH][q % CH];
    if (tid < CO)
        act_ws[(size_t)n * CO + tid] = act_o[tid];
}

// Spatial mean over the 64 positions of each batch element.
// d_out = [outputs (16*32)] ++ [pose_out (16*32*18)]
__global__ __launch_bounds__(256) void caps_reduce_kernel(
    const float* __restrict__ miu_ws, const float* __restrict__ act_ws,
    float* __restrict__ out)
{
    const int b = blockIdx.x;  // 0..15
    for (int item = threadIdx.x; item < CO + CO * CH; item += blockDim.x) {
        if (item < CO) {
            float s = 0.0f;
            for (int q = 0; q < 64; ++q) s += act_ws[(size_t)(b * 64 + q) * CO + item];
            out[b * CO + item] = s * (1.0f / 64.0f);
        } else {
            const int e = item - CO;  // 0..575 -> (c,ch)
            float s = 0.0f;
            for (int q = 0; q < 64; ++q) s += miu_ws[(size_t)(b * 64 + q) * (CO * CH) + e];
            out[16 * CO + b * (CO * CH) + e] = s * (1.0f / 64.0f);
        }
    }
}

extern "C" void kernel_launch(void* const* d_in, const int* in_sizes, int n_in,
                              void* d_out, int out_size, void* d_ws, size_t ws_size,
                              hipStream_t stream) {
    (void)in_sizes; (void)n_in; (void)out_size; (void)ws_size;
    const float* x        = (const float*)d_in[0];  // [16,8,8,544]
    const float* w        = (const float*)d_in[1];  // [1,32,32,4,4]
    const float* beta_v   = (const float*)d_in[2];  // [32,18]
    const float* beta_a   = (const float*)d_in[3];  // [32]
    const float* coord    = (const float*)d_in[4];  // [64,2]

    float* miu_ws = (float*)d_ws;                         // NPOS*32*18 floats
    float* act_ws = miu_ws + (size_t)NPOS * CO * CH;      // NPOS*32 floats

    caps_em_kernel<<<NPOS, 256, 0, stream>>>(x, w, beta_v, beta_a, coord,
                                             miu_ws, act_ws);
    caps_reduce_kernel<<<16, 256, 0, stream>>>(miu_ws, act_ws, (float*)d_out);
}